// NeuroFuzzyLayer_47047071760544
// MI455X (gfx1250) — compile-verified
//
#include <hip/hip_runtime.h>

// MI455X (gfx1250) fused NeuroFuzzy layer, fp32 WMMA (V_WMMA_F32_16X16X4_F32).
//
// Roofline: ~5.8 GFLOP fp32 vs ~25 MB HBM traffic -> compute bound; all math
// is mapped onto 16x16 WMMA tiles. Per 16-row tile (one wave32):
//   GEMM1: d2[16x32] = [z(x)z(upper-tri,2x) | z | 1][16x~] @ Bstat[~x32]
//          (quadratic-form expansion, S-symmetry halves the bilinear chunks)
//   softmax over k via 4x shfl_xor lane reductions (k lives on lanes in C-layout)
//   GEMM2: out[16x64] = (psi (x) y)[16x512] @ Wr[512x64] + psi[16x32] @ br[32x64]
// All B operands staged once per workgroup into ~191 KB of the 320 KB CDNA5
// WGP LDS in transposed, bank-conflict-free layouts; each WMMA fetches its B
// pair with one ds_load_b64. A-operand half-selection (K = 2h+{0,1}) is done
// by h-offset global b64 loads -- no per-lane register-array selects.

typedef __attribute__((ext_vector_type(2))) float v2f;
typedef __attribute__((ext_vector_type(8))) float v8f;

// LDS layout (float offsets)
#define BST_OFF 0             // Bstat^T : [32 k][stride 290], rows c<276 used
#define WRT_OFF 9280          // Wr^T    : [64 o][stride 514], c = k*16+d
#define BRT_OFF 42176         // br^T    : [64 o][stride 34],  c = k
#define PSI_OFF 44352         // per-wave psi scratch: 8 waves * 16 rows * stride 36
#define LDS_FLOATS 48960      // 195840 bytes

static __device__ __forceinline__ v8f wmma4(v2f a, v2f b, v8f c) {
  // D = A(16x4,f32) * B(4x16,f32) + C(16x16,f32)
  return __builtin_amdgcn_wmma_f32_16x16x4_f32(false, a, false, b, (short)0, c,
                                               false, false);
}

__global__ void __launch_bounds__(256, 1)
neurofuzzy_fused(const float* __restrict__ Y, const float* __restrict__ Z,
                 const float* __restrict__ MU, const float* __restrict__ SIG,
                 const float* __restrict__ W, const float* __restrict__ BIAS,
                 float* __restrict__ OUT) {
  extern __shared__ float lds[];
  const int tid = threadIdx.x;
  const int wid = tid >> 5;
  const int lane = tid & 31;
  const int M = lane & 15;   // row (A) or column (B/C) within 16
  const int h = lane >> 4;   // lane half: supplies K = 2h,2h+1 of each 4-chunk

  // ---------------- one-time per-workgroup staging ----------------
  // Wr^T[o][k*16+d] = W[(k*64+o)*16 + d]
  for (int idx = tid; idx < 64 * 512; idx += 256) {
    int o = idx >> 9, c = idx & 511, k = c >> 4, d = c & 15;
    lds[WRT_OFF + o * 514 + c] = W[(k * 64 + o) * 16 + d];
  }
  if (tid < 128) { int o = tid >> 1, p = tid & 1; lds[WRT_OFF + o * 514 + 512 + p] = 0.f; }
  // br^T[o][k] = BIAS[k*64+o]
  for (int idx = tid; idx < 2048; idx += 256) {
    int o = idx >> 5, k = idx & 31;
    lds[BRT_OFF + o * 34 + k] = BIAS[k * 64 + o];
  }
  if (tid < 128) { int o = tid >> 1, p = tid & 1; lds[BRT_OFF + o * 34 + 32 + p] = 0.f; }
  // Bstat^T[k][c]:
  //   c = d1*16+d2 < 256 : 0 (d2<d1) | S_k[d1][d1] (d2==d1) | 2*S_k[d1][d2] (d2>d1)
  //   256..271           : -2*(S_k mu_k)[d]
  //   272                : mu^T S mu
  {
    const int k = tid >> 3;
    const int dp = tid & 7;
    for (int s = 0; s < 2; ++s) {
      const int d1 = dp * 2 + s;
      float ar[16];
      const float* rp = SIG + (size_t)(k * 16 + d1) * 16;
#pragma unroll
      for (int m = 0; m < 16; ++m) ar[m] = rp[m];
      float srow[16];
      for (int e = 0; e < 16; ++e) {
        const float* ep = SIG + (size_t)(k * 16 + e) * 16;
        float s2 = 0.f;
#pragma unroll
        for (int m = 0; m < 16; ++m) s2 += ar[m] * ep[m];
        srow[e] = s2;  // true S_k[d1][e]
        float coeff = (e < d1) ? 0.f : ((e == d1) ? s2 : 2.f * s2);
        lds[BST_OFF + k * 290 + d1 * 16 + e] = coeff;
      }
      float q = 0.f;
      for (int e = 0; e < 16; ++e) q += srow[e] * MU[k * 16 + e];  // (S mu)_d1
      lds[BST_OFF + k * 290 + 256 + d1] = -2.f * q;
      lds[PSI_OFF + k * 16 + d1] = MU[k * 16 + d1] * q;  // partial of mu^T S mu
    }
  }
  __syncthreads();
  if (tid < 32) {
    const int k = tid;
    float cst = 0.f;
    for (int d = 0; d < 16; ++d) cst += lds[PSI_OFF + k * 16 + d];
    lds[BST_OFF + k * 290 + 272] = cst;
    for (int c = 273; c < 290; ++c) lds[BST_OFF + k * 290 + c] = 0.f;
  }
  __syncthreads();

  // ---------------- per-wave tile loop ----------------
  const int gw = blockIdx.x * 8 + wid;
  const int pbase = PSI_OFF + wid * 576;                 // this wave's psi scratch
  const int b1base0 = BST_OFF + M * 290 + 2 * h;         // d2 GEMM B, N-tile 0 (k 0..15)
  const int b1base1 = BST_OFF + (16 + M) * 290 + 2 * h;  // N-tile 1 (k 16..31)
  const int pArow = pbase + M * 36 + 2 * h;              // psi as A operand
  const int wb0 = WRT_OFF + M * 514 + 2 * h;
  const int wb1 = wb0 + 16 * 514;
  const int wb2 = wb0 + 32 * 514;
  const int wb3 = wb0 + 48 * 514;
  const int bb0 = BRT_OFF + M * 34 + 2 * h;
  const int bb1 = bb0 + 16 * 34;
  const int bb2 = bb0 + 32 * 34;
  const int bb3 = bb0 + 48 * 34;
  const int wstride = gridDim.x * 8;
  const float one_h = h ? 0.f : 1.f;  // A value for the constant column (c==272)

  for (int tile = gw; tile < 4096; tile += wstride) {
    const int rbase = tile * 16;
    const float* zrow = Z + (size_t)(rbase + M) * 16;
    const float* yrow = Y + (size_t)(rbase + M) * 16;

    // full z row (multipliers z_d1), global_load_b128 x4
    float zr[16];
#pragma unroll
    for (int q4 = 0; q4 < 4; ++q4) {
      float4 t = ((const float4*)zrow)[q4];
      zr[q4 * 4 + 0] = t.x; zr[q4 * 4 + 1] = t.y;
      zr[q4 * 4 + 2] = t.z; zr[q4 * 4 + 3] = t.w;
    }
    // half-selected operand pairs zs[2m+i] = z[4m+2h+i], ys likewise:
    // loaded directly with h-offset b64 loads (no register selects).
    float zs[8], ys[8];
#pragma unroll
    for (int m = 0; m < 4; ++m) {
      float2 tz = *(const float2*)(zrow + 4 * m + 2 * h);
      zs[2 * m + 0] = tz.x; zs[2 * m + 1] = tz.y;
      float2 ty = *(const float2*)(yrow + 4 * m + 2 * h);
      ys[2 * m + 0] = ty.x; ys[2 * m + 1] = ty.y;
    }
    {  // uniform, branchless prefetch of the next tile's rows
      int pnext = tile + wstride;
      pnext = (pnext < 4096) ? pnext : tile;
      __builtin_prefetch(Z + (size_t)(pnext * 16 + M) * 16, 0, 3);
      __builtin_prefetch(Y + (size_t)(pnext * 16 + M) * 16, 0, 3);
    }

    // ---- GEMM1: d2[16x32], upper-triangular quadratic + linear + const ----
    v8f d2a = {0, 0, 0, 0, 0, 0, 0, 0};
    v8f d2b = {0, 0, 0, 0, 0, 0, 0, 0};
#pragma unroll
    for (int d1 = 0; d1 < 16; ++d1) {
#pragma unroll
      for (int m = 0; m < 4; ++m) {
        if (4 * m + 3 < d1) continue;  // all-zero chunk (lower triangle), pruned at compile time
        const int j = d1 * 4 + m;
        v2f a;
        a.x = zr[d1] * zs[2 * m + 0];
        a.y = zr[d1] * zs[2 * m + 1];
        v2f b0 = *(const v2f*)&lds[b1base0 + 4 * j];
        v2f b1 = *(const v2f*)&lds[b1base1 + 4 * j];
        d2a = wmma4(a, b0, d2a);
        d2b = wmma4(a, b1, d2b);
      }
    }
#pragma unroll
    for (int m = 0; m < 4; ++m) {  // linear region: A = z_d
      const int j = 64 + m;
      v2f a;
      a.x = zs[2 * m + 0];
      a.y = zs[2 * m + 1];
      v2f b0 = *(const v2f*)&lds[b1base0 + 4 * j];
      v2f b1 = *(const v2f*)&lds[b1base1 + 4 * j];
      d2a = wmma4(a, b0, d2a);
      d2b = wmma4(a, b1, d2b);
    }
    {  // constant column (c==272)
      v2f a;
      a.x = one_h;
      a.y = 0.f;
      v2f b0 = *(const v2f*)&lds[b1base0 + 4 * 68];
      v2f b1 = *(const v2f*)&lds[b1base1 + 4 * 68];
      d2a = wmma4(a, b0, d2a);
      d2b = wmma4(a, b1, d2b);
    }

    // ---- softmax over k (clamp like reference), k on lanes in C-layout ----
    float psa[8], psb[8];
#pragma unroll
    for (int v = 0; v < 8; ++v) {
      float da = fmaxf(d2a[v], 1e-12f);
      float db = fmaxf(d2b[v], 1e-12f);
      float mn = fminf(da, db);
      mn = fminf(mn, __shfl_xor(mn, 1));
      mn = fminf(mn, __shfl_xor(mn, 2));
      mn = fminf(mn, __shfl_xor(mn, 4));
      mn = fminf(mn, __shfl_xor(mn, 8));
      float ea = __expf(mn - da);
      float eb = __expf(mn - db);
      float s = ea + eb;
      s += __shfl_xor(s, 1);
      s += __shfl_xor(s, 2);
      s += __shfl_xor(s, 4);
      s += __shfl_xor(s, 8);
      float inv = 1.f / s;
      psa[v] = ea * inv;
      psb[v] = eb * inv;
    }
    // transpose psi through LDS: psi[row][k], row stride 36 (conflict-free)
#pragma unroll
    for (int v = 0; v < 8; ++v) {
      lds[pbase + (v + 8 * h) * 36 + M] = psa[v];
      lds[pbase + (v + 8 * h) * 36 + 16 + M] = psb[v];
    }
    asm volatile("s_wait_dscnt 0x0" ::: "memory");

    // ---- GEMM2: out[16x64] = psi@br + (psi (x) y) @ Wr ----
    v8f o0 = {0, 0, 0, 0, 0, 0, 0, 0};
    v8f o1 = {0, 0, 0, 0, 0, 0, 0, 0};
    v8f o2 = {0, 0, 0, 0, 0, 0, 0, 0};
    v8f o3 = {0, 0, 0, 0, 0, 0, 0, 0};
#pragma unroll
    for (int j = 0; j < 8; ++j) {  // bias GEMM, K = 32, A = psi
      v2f a = *(const v2f*)&lds[pArow + 4 * j];
      o0 = wmma4(a, *(const v2f*)&lds[bb0 + 4 * j], o0);
      o1 = wmma4(a, *(const v2f*)&lds[bb1 + 4 * j], o1);
      o2 = wmma4(a, *(const v2f*)&lds[bb2 + 4 * j], o2);
      o3 = wmma4(a, *(const v2f*)&lds[bb3 + 4 * j], o3);
    }
    for (int kk = 0; kk < 32; ++kk) {  // main GEMM, K = 512, A = psi_k * y_d
      float pk = lds[pbase + M * 36 + kk];
#pragma unroll
      for (int jj = 0; jj < 4; ++jj) {
        const int j = kk * 4 + jj;
        v2f a;
        a.x = pk * ys[2 * jj + 0];
        a.y = pk * ys[2 * jj + 1];
        o0 = wmma4(a, *(const v2f*)&lds[wb0 + 4 * j], o0);
        o1 = wmma4(a, *(const v2f*)&lds[wb1 + 4 * j], o1);
        o2 = wmma4(a, *(const v2f*)&lds[wb2 + 4 * j], o2);
        o3 = wmma4(a, *(const v2f*)&lds[wb3 + 4 * j], o3);
      }
    }

    // ---- store out[16 rows x 64 o] (C-layout: row = v+8h, o = 16t+M) ----
    float* outp = OUT + (size_t)rbase * 64;
#pragma unroll
    for (int v = 0; v < 8; ++v) {
      const int r = v + 8 * h;
      outp[r * 64 + 0 + M] = o0[v];
      outp[r * 64 + 16 + M] = o1[v];
      outp[r * 64 + 32 + M] = o2[v];
      outp[r * 64 + 48 + M] = o3[v];
    }
  }
}

extern "C" void kernel_launch(void* const* d_in, const int* in_sizes, int n_in,
                              void* d_out, int out_size, void* d_ws, size_t ws_size,
                              hipStream_t stream) {
  (void)in_sizes; (void)n_in; (void)d_ws; (void)ws_size; (void)out_size;
  const float* Y = (const float*)d_in[0];
  const float* Z = (const float*)d_in[1];
  const float* MU = (const float*)d_in[2];
  const float* SIG = (const float*)d_in[3];
  const float* W = (const float*)d_in[4];
  const float* BIAS = (const float*)d_in[5];
  float* OUT = (float*)d_out;

  const size_t shmem = (size_t)LDS_FLOATS * sizeof(float);  // 195840 B < 320 KB WGP LDS
  (void)hipFuncSetAttribute((const void*)neurofuzzy_fused,
                            hipFuncAttributeMaxDynamicSharedMemorySize, (int)shmem);
  hipLaunchKernelGGL(neurofuzzy_fused, dim3(128), dim3(256), shmem, stream,
                     Y, Z, MU, SIG, W, BIAS, OUT);
}